// RFMCNN_31361851196149
// MI455X (gfx1250) — compile-verified
//
#include <hip/hip_runtime.h>
#include <math.h>

#define N     4096
#define D2    1024
#define NRHS  10
#define RS    16        // padded RHS stride
#define NB    64        // Cholesky block / K width
#define MT    128       // trailing-update block tile (MT x MT)
#define LP    68        // LDS row padding (bank-conflict-free fragment reads)
#define INV_L 0.1f      // 1/BANDWIDTH
#define REGL  1e-3f

typedef __attribute__((ext_vector_type(2))) float v2f;
typedef __attribute__((ext_vector_type(8))) float v8f;

// ---------------------------------------------------------------- m = mean(relu(conv3x3(X)))
__global__ void k_mean_feat(const float* __restrict__ X, const float* __restrict__ W,
                            float* __restrict__ m) {
    __shared__ float red[256];
    int pix = blockIdx.x;               // 0..1023
    int y = pix >> 5, x = pix & 31;
    float w[9];
#pragma unroll
    for (int t = 0; t < 9; ++t) w[t] = W[t];
    float acc = 0.f;
    for (int i = threadIdx.x; i < N; i += 256) {
        const float* row = X + (size_t)i * D2;
        float s = 0.f;
#pragma unroll
        for (int ky = 0; ky < 3; ++ky) {
            int yy = y + ky - 1;
            if (yy < 0 || yy >= 32) continue;
#pragma unroll
            for (int kx = 0; kx < 3; ++kx) {
                int xx = x + kx - 1;
                if (xx < 0 || xx >= 32) continue;
                s += w[ky * 3 + kx] * row[yy * 32 + xx];
            }
        }
        acc += fmaxf(s, 0.f);
    }
    red[threadIdx.x] = acc;
    __syncthreads();
    for (int off = 128; off > 0; off >>= 1) {
        if (threadIdx.x < off) red[threadIdx.x] += red[threadIdx.x + off];
        __syncthreads();
    }
    if (threadIdx.x == 0) m[pix] = red[0] * (1.0f / N);
}

// ---------------------------------------------------------------- p = X @ m
__global__ void k_dot_p(const float* __restrict__ X, const float* __restrict__ m,
                        float* __restrict__ p) {
    __shared__ float red[256];
    int i = blockIdx.x;
    const float* row = X + (size_t)i * D2;
    float s = 0.f;
    for (int k = threadIdx.x; k < D2; k += 256) s += row[k] * m[k];
    red[threadIdx.x] = s;
    __syncthreads();
    for (int off = 128; off > 0; off >>= 1) {
        if (threadIdx.x < off) red[threadIdx.x] += red[threadIdx.x + off];
        __syncthreads();
    }
    if (threadIdx.x == 0) p[i] = red[0];
}

// ---------------------------------------------------------------- A[i,j] = exp(-|pi-pj|/L) + reg*I
__global__ void k_build_A(const float* __restrict__ p, float* __restrict__ A) {
    int i = blockIdx.y;
    int j = blockIdx.x * 256 + threadIdx.x;
    float d = fabsf(p[i] - p[j]);
    float v = expf(-d * INV_L);
    if (i == j) v += REGL;
    A[(size_t)i * N + j] = v;
}

// ---------------------------------------------------------------- 64x64 diagonal Cholesky (64 threads)
__global__ void k_chol_diag(float* __restrict__ A, int k0) {
    __shared__ float T[NB][NB + 1];
    int t = threadIdx.x;                // 0..63, owns row t
    for (int j = 0; j < NB; ++j) T[t][j] = A[(size_t)(k0 + t) * N + k0 + j];
    __syncthreads();
    for (int j = 0; j < NB; ++j) {
        if (t == j) T[j][j] = sqrtf(T[j][j]);
        __syncthreads();
        if (t > j) T[t][j] /= T[j][j];
        __syncthreads();
        if (t > j) {
            float ltj = T[t][j];
            for (int jj = j + 1; jj <= t; ++jj) T[t][jj] -= ltj * T[jj][j];
        }
        __syncthreads();
    }
    for (int j = 0; j < NB; ++j) A[(size_t)(k0 + t) * N + k0 + j] = T[t][j];
}

// ---------------------------------------------------------------- panel rows: x * L^T = a (forward subst per row)
__global__ void k_panel_trsm(float* __restrict__ A, int k0) {
    __shared__ float Ld[NB][NB + 1];
    __shared__ float xs[NB][128];
    int tid = threadIdx.x;              // 128 threads
    for (int idx = tid; idx < NB * NB; idx += 128) {
        int r = idx >> 6, c = idx & 63;
        Ld[r][c] = A[(size_t)(k0 + r) * N + k0 + c];
    }
    __syncthreads();
    int i = k0 + NB + blockIdx.x * 128 + tid;
    if (i < N) {
        float* row = A + (size_t)i * N + k0;
        for (int j = 0; j < NB; ++j) {
            float s = row[j];
            for (int jj = 0; jj < j; ++jj) s -= xs[jj][tid] * Ld[j][jj];
            s /= Ld[j][j];
            xs[j][tid] = s;
            row[j] = s;
        }
    }
}

// ---------------------------------------------------------------- trailing update: C -= Pi * Pj^T (fp32 WMMA)
// 256 threads = 8 waves; block tile MTxMT; wave w: rows (w&3)*32..+32, cols (w>>2)*64..+64
__global__ __launch_bounds__(256) void k_syrk_update(float* __restrict__ A, int k0) {
    __shared__ float Pi[MT][LP];
    __shared__ float Pj[MT][LP];
    int i0 = k0 + NB + blockIdx.y * MT;
    int j0 = k0 + NB + blockIdx.x * MT;
    int tid = threadIdx.x;

    // async-stage both MTx64 panels: 2048 16B chunks each, 8 per thread per panel
    const char* Abase = (const char*)A;
#pragma unroll
    for (int it = 0; it < 8; ++it) {
        int ch = tid + it * 256;
        int r  = ch >> 4;               // 0..127
        int c4 = (ch & 15) << 2;        // 0,4,..,60
        int ri = i0 + r; if (ri > N - 1) ri = N - 1;   // clamp: safe reads at edges
        int rj = j0 + r; if (rj > N - 1) rj = N - 1;
        unsigned goff_i = ((unsigned)ri * (unsigned)N + (unsigned)(k0 + c4)) * 4u;
        unsigned goff_j = ((unsigned)rj * (unsigned)N + (unsigned)(k0 + c4)) * 4u;
        unsigned lds_i = (unsigned)(unsigned long long)(&Pi[r][c4]);
        unsigned lds_j = (unsigned)(unsigned long long)(&Pj[r][c4]);
        asm volatile("global_load_async_to_lds_b128 %0, %1, %2"
                     :: "v"(lds_i), "v"(goff_i), "s"(Abase) : "memory");
        asm volatile("global_load_async_to_lds_b128 %0, %1, %2"
                     :: "v"(lds_j), "v"(goff_j), "s"(Abase) : "memory");
    }
    asm volatile("s_wait_asynccnt 0x0" ::: "memory");
    __syncthreads();

    int lane  = tid & 31;
    int wave  = tid >> 5;
    int lr    = lane & 15;
    int khalf = (lane >> 4) * 2;        // lanes 0-15: K={0,1}; lanes 16-31: K={2,3}
    int rb    = (wave & 3) * 32;
    int cb    = (wave >> 2) * 64;
    v8f acc[2][4] = {};
    for (int kk = 0; kk < NB; kk += 4) {
        v2f a0, a1, b[4];
        a0.x = Pi[rb + lr][kk + khalf];      a0.y = Pi[rb + lr][kk + khalf + 1];
        a1.x = Pi[rb + 16 + lr][kk + khalf]; a1.y = Pi[rb + 16 + lr][kk + khalf + 1];
#pragma unroll
        for (int t = 0; t < 4; ++t) {
            b[t].x = Pj[cb + t * 16 + lr][kk + khalf];
            b[t].y = Pj[cb + t * 16 + lr][kk + khalf + 1];
        }
#pragma unroll
        for (int t = 0; t < 4; ++t) {
            acc[0][t] = __builtin_amdgcn_wmma_f32_16x16x4_f32(false, a0, false, b[t],
                                                              (short)0, acc[0][t], false, false);
            acc[1][t] = __builtin_amdgcn_wmma_f32_16x16x4_f32(false, a1, false, b[t],
                                                              (short)0, acc[1][t], false, false);
        }
    }
    // C/D layout: lane = n + 16*(m>>3), vgpr r -> m = r + 8*(lane>>4)
    int rsel = (lane >> 4) * 8;
    int col  = lane & 15;
#pragma unroll
    for (int R = 0; R < 2; ++R) {
#pragma unroll
        for (int t = 0; t < 4; ++t) {
#pragma unroll
            for (int r = 0; r < 8; ++r) {
                int gi = i0 + rb + R * 16 + rsel + r;
                int gj = j0 + cb + t * 16 + col;
                if (gi < N && gj < N) {
                    size_t off = (size_t)gi * N + gj;
                    A[off] -= acc[R][t][r];
                }
            }
        }
    }
}

// ---------------------------------------------------------------- RHS init: Z[i][c] = y[i][c]
__global__ void k_zinit(const float* __restrict__ y, float* __restrict__ Z) {
    int idx = blockIdx.x * 256 + threadIdx.x;
    if (idx < N * NRHS) {
        int i = idx / NRHS, c = idx % NRHS;
        Z[i * RS + c] = y[idx];
    }
}

// ---------------------------------------------------------------- forward solve: diag block
__global__ void k_fwd_diag(const float* __restrict__ A, float* __restrict__ Z, int k0) {
    __shared__ float Ld[NB][NB + 1];
    __shared__ float zb[NB][NRHS + 1];
    int tid = threadIdx.x;              // 256
    for (int idx = tid; idx < NB * NB; idx += 256) {
        int r = idx >> 6, c = idx & 63;
        Ld[r][c] = A[(size_t)(k0 + r) * N + k0 + c];
    }
    for (int idx = tid; idx < NB * NRHS; idx += 256) {
        int r = idx / NRHS, c = idx % NRHS;
        zb[r][c] = Z[(k0 + r) * RS + c];
    }
    __syncthreads();
    for (int r = 0; r < NB; ++r) {
        if (tid < NRHS) zb[r][tid] /= Ld[r][r];
        __syncthreads();
        int cnt = (NB - 1 - r) * NRHS;
        for (int idx = tid; idx < cnt; idx += 256) {
            int rr = r + 1 + idx / NRHS, c = idx % NRHS;
            zb[rr][c] -= Ld[rr][r] * zb[r][c];
        }
        __syncthreads();
    }
    for (int idx = tid; idx < NB * NRHS; idx += 256) {
        int r = idx / NRHS, c = idx % NRHS;
        Z[(k0 + r) * RS + c] = zb[r][c];
    }
}

// ---------------------------------------------------------------- forward solve: trailing RHS update
__global__ void k_fwd_update(const float* __restrict__ A, float* __restrict__ Z, int k0) {
    __shared__ float zs[NB][NRHS];
    int tid = threadIdx.x;              // 256
    for (int idx = tid; idx < NB * NRHS; idx += 256) {
        int r = idx / NRHS, c = idx % NRHS;
        zs[r][c] = Z[(k0 + r) * RS + c];
    }
    __syncthreads();
    int i = k0 + NB + blockIdx.x * 256 + tid;
    if (i < N) {
        const float* Lrow = A + (size_t)i * N + k0;
        float acc[NRHS];
#pragma unroll
        for (int c = 0; c < NRHS; ++c) acc[c] = 0.f;
        for (int j = 0; j < NB; ++j) {
            float lv = Lrow[j];
#pragma unroll
            for (int c = 0; c < NRHS; ++c) acc[c] += lv * zs[j][c];
        }
#pragma unroll
        for (int c = 0; c < NRHS; ++c) Z[i * RS + c] -= acc[c];
    }
}

// ---------------------------------------------------------------- backward solve (L^T): diag block
__global__ void k_bwd_diag(const float* __restrict__ A, float* __restrict__ Z, int k0) {
    __shared__ float Ld[NB][NB + 1];
    __shared__ float zb[NB][NRHS + 1];
    int tid = threadIdx.x;              // 256
    for (int idx = tid; idx < NB * NB; idx += 256) {
        int r = idx >> 6, c = idx & 63;
        Ld[r][c] = A[(size_t)(k0 + r) * N + k0 + c];
    }
    for (int idx = tid; idx < NB * NRHS; idx += 256) {
        int r = idx / NRHS, c = idx % NRHS;
        zb[r][c] = Z[(k0 + r) * RS + c];
    }
    __syncthreads();
    for (int r = NB - 1; r >= 0; --r) {
        if (tid < NRHS) zb[r][tid] /= Ld[r][r];
        __syncthreads();
        int cnt = r * NRHS;
        for (int idx = tid; idx < cnt; idx += 256) {
            int rr = idx / NRHS, c = idx % NRHS;
            zb[rr][c] -= Ld[r][rr] * zb[r][c];   // L^T[rr][r] = L[r][rr]
        }
        __syncthreads();
    }
    for (int idx = tid; idx < NB * NRHS; idx += 256) {
        int r = idx / NRHS, c = idx % NRHS;
        Z[(k0 + r) * RS + c] = zb[r][c];
    }
}

// ---------------------------------------------------------------- backward solve: RHS update above block
__global__ void k_bwd_update(const float* __restrict__ A, float* __restrict__ Z, int k0) {
    __shared__ float zs[NB][NRHS];
    int tid = threadIdx.x;              // 256
    for (int idx = tid; idx < NB * NRHS; idx += 256) {
        int r = idx / NRHS, c = idx % NRHS;
        zs[r][c] = Z[(k0 + r) * RS + c];
    }
    __syncthreads();
    int i = blockIdx.x * 256 + tid;     // rows above block
    if (i < k0) {
        float acc[NRHS];
#pragma unroll
        for (int c = 0; c < NRHS; ++c) acc[c] = 0.f;
        for (int j = 0; j < NB; ++j) {
            float lv = A[(size_t)(k0 + j) * N + i];   // L[k0+j][i]
#pragma unroll
            for (int c = 0; c < NRHS; ++c) acc[c] += lv * zs[j][c];
        }
#pragma unroll
        for (int c = 0; c < NRHS; ++c) Z[i * RS + c] -= acc[c];
    }
}

// ---------------------------------------------------------------- out[c][i] = Z[i][c]
__global__ void k_writeout(const float* __restrict__ Z, float* __restrict__ out) {
    int i = blockIdx.x * 256 + threadIdx.x;
    if (i < N) {
#pragma unroll
        for (int c = 0; c < NRHS; ++c) out[c * N + i] = Z[i * RS + c];
    }
}

extern "C" void kernel_launch(void* const* d_in, const int* in_sizes, int n_in,
                              void* d_out, int out_size, void* d_ws, size_t ws_size,
                              hipStream_t stream) {
    (void)in_sizes; (void)n_in; (void)out_size; (void)ws_size;
    const float* X = (const float*)d_in[0];
    const float* y = (const float*)d_in[1];
    const float* W = (const float*)d_in[2];
    float* out = (float*)d_out;

    float* A = (float*)d_ws;                    // N*N
    float* Z = A + (size_t)N * N;               // N*RS
    float* m = Z + (size_t)N * RS;              // D2
    float* p = m + D2;                          // N

    // rank-1 collapse: m, p, then A = K + reg*I
    k_mean_feat<<<D2, 256, 0, stream>>>(X, W, m);
    k_dot_p<<<N, 256, 0, stream>>>(X, m, p);
    dim3 gA(N / 256, N);
    k_build_A<<<gA, 256, 0, stream>>>(p, A);

    // blocked Cholesky
    for (int kb = 0; kb < N / NB; ++kb) {
        int k0 = kb * NB;
        k_chol_diag<<<1, NB, 0, stream>>>(A, k0);
        int rem = N - k0 - NB;
        if (rem > 0) {
            k_panel_trsm<<<(rem + 127) / 128, 128, 0, stream>>>(A, k0);
            int g = (rem + MT - 1) / MT;
            dim3 gg(g, g);
            k_syrk_update<<<gg, 256, 0, stream>>>(A, k0);
        }
    }

    // triangular solves, 10 RHS
    k_zinit<<<(N * NRHS + 255) / 256, 256, 0, stream>>>(y, Z);
    for (int kb = 0; kb < N / NB; ++kb) {
        int k0 = kb * NB;
        k_fwd_diag<<<1, 256, 0, stream>>>(A, Z, k0);
        int rem = N - k0 - NB;
        if (rem > 0) k_fwd_update<<<(rem + 255) / 256, 256, 0, stream>>>(A, Z, k0);
    }
    for (int kb = N / NB - 1; kb >= 0; --kb) {
        int k0 = kb * NB;
        k_bwd_diag<<<1, 256, 0, stream>>>(A, Z, k0);
        if (k0 > 0) k_bwd_update<<<(k0 + 255) / 256, 256, 0, stream>>>(A, Z, k0);
    }
    k_writeout<<<(N + 255) / 256, 256, 0, stream>>>(Z, out);
}